// GNN_LSTM_75539884802671
// MI455X (gfx1250) — compile-verified
//
#include <hip/hip_runtime.h>
#include <math.h>
#include <stdint.h>

// ---------------- problem constants ----------------
#define TT    96
#define NN    1024
#define GG    128
#define HH    512
#define EE    32768
#define D2    56
#define INW   64
#define ETOT  (EE + NN)          // edges + self loops = 33792

typedef __attribute__((ext_vector_type(2))) float v2f;
typedef __attribute__((ext_vector_type(8))) float v8f;

// ---------------- CDNA5 async global->LDS path (guarded) ----------------
#if defined(__has_builtin)
#if __has_builtin(__builtin_amdgcn_global_load_async_to_lds_b128) && \
    __has_builtin(__builtin_amdgcn_s_wait_asynccnt)
#define USE_ASYNC_LDS 1
#endif
#endif

#ifdef USE_ASYNC_LDS
#pragma message("CDNA5: using global_load_async_to_lds_b128 double-buffer path")
// builtin expects: (v4i addrspace(1)*, v4i addrspace(3)*, imm offset, imm cpol)
typedef int v4i __attribute__((vector_size(4 * sizeof(int))));
typedef __attribute__((address_space(1))) v4i* as1_v4i;
typedef __attribute__((address_space(3))) v4i* as3_v4i;
__device__ __forceinline__ void cdna5_async_copy16(const float* g, float* l) {
  // flat LDS address truncates to the 32-bit LDS offset (ISA 10.2 aperture rule)
  __builtin_amdgcn_global_load_async_to_lds_b128(
      (as1_v4i)(uintptr_t)g, (as3_v4i)(uint32_t)(uintptr_t)l, 0, 0);
}
#else
#pragma message("CDNA5: async builtins unavailable; synchronous LDS staging")
#endif

// ---------------- zero fill ----------------
__global__ __launch_bounds__(256) void gl_zero_f32(float* __restrict__ p, int n) {
  int i = blockIdx.x * 256 + threadIdx.x;
  if (i < n) p[i] = 0.0f;
}

// ---------------- degree count (atomic) ----------------
__global__ __launch_bounds__(256) void gl_deg_count(const int* __restrict__ ei,
                                                    float* __restrict__ deg) {
  int t = blockIdx.y;
  int e = blockIdx.x * 256 + threadIdx.x;
  if (e < EE) {
    int dst = ei[(long long)t * 2 * EE + EE + e];
    atomicAdd(&deg[t * NN + dst], 1.0f);
  }
}

// deg -> dinv  (every node has a self loop, so deg+1 >= 1)
__global__ __launch_bounds__(256) void gl_dinv(float* __restrict__ deg) {
  int i = blockIdx.x * 256 + threadIdx.x;
  if (i < TT * NN) deg[i] = rsqrtf(deg[i] + 1.0f);
}

// ---------------- WMMA fp32 GEMM: C[1024x128] = A[1024xK] @ B[Kx128], batched over T ----
// block = 256 threads (8 waves). Each wave owns a 16-row strip, all 128 cols.
// Double-buffered LDS tiles, staged with async-to-LDS when available.
#define LDA_S 36      // A tile LDS stride (row-major [row][k]); 36*4B = 144B -> 16B aligned rows
#define LDB_S 136     // B tile LDS stride (row-major [k][n]);   136%32==8 -> frag halves disjoint

__global__ __launch_bounds__(256) void gl_gemm_wmma(
    const float* __restrict__ Ab, long long strideA, int lda,
    const float* __restrict__ B,
    float* __restrict__ Cb, long long strideC, int K) {
  __shared__ float As[2][128 * LDA_S];   // 2 x (128 rows x 32 k)
  __shared__ float Bs[2][32 * LDB_S];    // 2 x (32 k x 128 n)

  const int t   = blockIdx.y;
  const float* A = Ab + (long long)t * strideA + (long long)blockIdx.x * 128 * lda;
  float*       C = Cb + (long long)t * strideC + (long long)blockIdx.x * 128 * GG;

  const int tid  = threadIdx.x;
  const int wave = tid >> 5;        // 0..7
  const int lane = tid & 31;
  const int m    = lane & 15;
  const int hi   = lane >> 4;       // 0/1 : which K-half of the fragment

  // staging coordinates for this thread
  const int kq = tid & 7;           // A: k quad 0..7 -> k = 4*kq
  const int r0 = tid >> 3;          // A: row 0..31 (4 passes of +32)
  const int bcol = (tid & 31) * 4;  // B: col
  const int bk0  = tid >> 5;        // B: k 0..7 (4 passes of +8)

  v8f acc[8];
  {
    v8f z = {0.f, 0.f, 0.f, 0.f, 0.f, 0.f, 0.f, 0.f};
    for (int i = 0; i < 8; i++) acc[i] = z;
  }

  const int nc = K >> 5;            // number of 32-wide k chunks

  // ---- stage one chunk into buffer `buf` ----
  auto stage = [&](int buf, int kc) {
    for (int p = 0; p < 4; p++) {
      int row = r0 + 32 * p;
      const float* gA = A + (long long)row * lda + kc + 4 * kq;
      float* lA = &As[buf][row * LDA_S + 4 * kq];
#ifdef USE_ASYNC_LDS
      cdna5_async_copy16(gA, lA);
#else
      *reinterpret_cast<float4*>(lA) = *reinterpret_cast<const float4*>(gA);
#endif
    }
    for (int p = 0; p < 4; p++) {
      int k = bk0 + 8 * p;
      const float* gB = B + (long long)(kc + k) * GG + bcol;
      float* lB = &Bs[buf][k * LDB_S + bcol];
#ifdef USE_ASYNC_LDS
      cdna5_async_copy16(gB, lB);
#else
      *reinterpret_cast<float4*>(lB) = *reinterpret_cast<const float4*>(gB);
#endif
    }
  };

  stage(0, 0);                      // prologue

  for (int c = 0; c < nc; c++) {
    const int buf = c & 1;
    if (c + 1 < nc) stage(buf ^ 1, (c + 1) * 32);   // prefetch next chunk
#ifdef USE_ASYNC_LDS
    // 8 async ops just issued for the next stage may remain outstanding;
    // waiting to <=8 proves the previous stage's 8 ops completed (in-order).
    if (c + 1 < nc) __builtin_amdgcn_s_wait_asynccnt(8);
    else            __builtin_amdgcn_s_wait_asynccnt(0);
#endif
    __syncthreads();                // all waves' staging visible

    // --- 8 wmma k-steps of 4, for 8 col tiles ---
    for (int kk = 0; kk < 8; kk++) {
      int k0 = 4 * kk;
      v2f a;
      int arow = wave * 16 + m;
      a.x = As[buf][arow * LDA_S + k0 + 2 * hi];        // A[m][k0+2hi]
      a.y = As[buf][arow * LDA_S + k0 + 2 * hi + 1];    // A[m][k0+2hi+1]
      for (int nt = 0; nt < 8; nt++) {
        v2f b;
        b.x = Bs[buf][(k0 + 2 * hi) * LDB_S + nt * 16 + m];       // B[k0+2hi][n]
        b.y = Bs[buf][(k0 + 2 * hi + 1) * LDB_S + nt * 16 + m];   // B[k0+2hi+1][n]
        acc[nt] = __builtin_amdgcn_wmma_f32_16x16x4_f32(
            false, a, false, b, (short)0, acc[nt], false, false);
      }
    }
    __syncthreads();                // readers done before buffer reuse
  }

  // --- store C: VGPR v holds row (v + 8*hi), col = nt*16 + m ---
  for (int nt = 0; nt < 8; nt++)
    for (int v = 0; v < 8; v++) {
      int row = wave * 16 + v + 8 * hi;
      C[(long long)row * GG + nt * 16 + m] = acc[nt][v];
    }
}

// ---------------- edge scatter: out[dst] += xw[src] * dinv[src]*dinv[dst] ----------------
// one wave per edge, lane handles 4 of 128 features
__global__ __launch_bounds__(256) void gl_scatter(const int* __restrict__ ei,
                                                  const float* __restrict__ dinv,
                                                  const float* __restrict__ xw,
                                                  float* __restrict__ out) {
  int t = blockIdx.y;
  long long eidx = (long long)blockIdx.x * 8 + (threadIdx.x >> 5);
  int lane = threadIdx.x & 31;
  if (eidx >= ETOT) return;
  int src, dst;
  if (eidx < EE) {
    src = ei[(long long)t * 2 * EE + eidx];
    dst = ei[(long long)t * 2 * EE + EE + eidx];
  } else {
    src = dst = (int)(eidx - EE);     // self loop
  }
  float nrm = dinv[t * NN + src] * dinv[t * NN + dst];
  const float* srow = xw  + ((long long)t * NN + src) * GG;
  float*       drow = out + ((long long)t * NN + dst) * GG;
  float4 v = *reinterpret_cast<const float4*>(srow + lane * 4);
  atomicAdd(drow + lane * 4 + 0, v.x * nrm);
  atomicAdd(drow + lane * 4 + 1, v.y * nrm);
  atomicAdd(drow + lane * 4 + 2, v.z * nrm);
  atomicAdd(drow + lane * 4 + 3, v.w * nrm);
}

// ---------------- bias (+ optional relu), in place over [T*N, 128] ----------------
__global__ __launch_bounds__(256) void gl_bias_act(float* __restrict__ x,
                                                   const float* __restrict__ b, int relu) {
  long long i = (long long)blockIdx.x * 256 + threadIdx.x;
  if (i < (long long)TT * NN * GG) {
    int c = (int)(i & (GG - 1));
    float v = x[i] + b[c];
    if (relu) v = fmaxf(v, 0.0f);
    x[i] = v;
  }
}

// ---------------- FC: gtmp[t][j] += sum_i xh2[t][i] * Wfc[i][j] ----------------
// grid (T, 16 chunks of 8192), 224 threads = 56 j x 4 i-subsets
__global__ __launch_bounds__(224) void gl_fc(const float* __restrict__ xh2,
                                             const float* __restrict__ Wfc,
                                             float* __restrict__ gtmp) {
  __shared__ float xs[1024];
  __shared__ float red[4][D2];
  int t = blockIdx.x, chunk = blockIdx.y;
  int tid = threadIdx.x;
  int s = tid / D2, j = tid % D2;
  const float* xrow = xh2 + (long long)t * (NN * GG) + (long long)chunk * 8192;
  float acc = 0.0f;
  for (int base = 0; base < 8192; base += 1024) {
    __syncthreads();
    for (int i = tid; i < 1024; i += 224) xs[i] = xrow[base + i];
    __syncthreads();
    const float* wp = Wfc + (long long)(chunk * 8192 + base) * D2 + j;
    for (int ii = s; ii < 1024; ii += 4) acc += xs[ii] * wp[(long long)ii * D2];
  }
  red[s][j] = acc;
  __syncthreads();
  if (s == 0) {
    float sum = red[0][j] + red[1][j] + red[2][j] + red[3][j];
    atomicAdd(&gtmp[t * D2 + j], sum);
  }
}

__global__ __launch_bounds__(64) void gl_g_act(const float* __restrict__ gtmp,
                                               const float* __restrict__ bfc,
                                               float* __restrict__ g) {
  int i = blockIdx.x * 64 + threadIdx.x;
  if (i < TT * D2) g[i] = fmaxf(gtmp[i] + bfc[i % D2], 0.0f);
}

// ---------------- pre-gates: pre[t][gate][h] = x_t@W_i + g@W_g + b ----------------
__global__ __launch_bounds__(512) void gl_pregate(
    const float* __restrict__ inp, const float* __restrict__ g,
    const float* __restrict__ W_ii, const float* __restrict__ W_gi, const float* __restrict__ b_i,
    const float* __restrict__ W_if, const float* __restrict__ W_gf, const float* __restrict__ b_f,
    const float* __restrict__ W_ig, const float* __restrict__ W_gg, const float* __restrict__ b_g,
    const float* __restrict__ W_io, const float* __restrict__ W_go, const float* __restrict__ b_o,
    float* __restrict__ pre) {
  int t = blockIdx.x, gate = blockIdx.y, h = threadIdx.x;
  const float *Wi, *Wg, *b;
  if      (gate == 0) { Wi = W_ii; Wg = W_gi; b = b_i; }
  else if (gate == 1) { Wi = W_if; Wg = W_gf; b = b_f; }
  else if (gate == 2) { Wi = W_ig; Wg = W_gg; b = b_g; }
  else                { Wi = W_io; Wg = W_go; b = b_o; }
  float acc = b[h];
  const float* x = inp + (long long)t * INW;
  for (int k = 0; k < INW; k++) acc += x[k] * Wi[(long long)k * HH + h];
  const float* gv = g + t * D2;
  for (int k = 0; k < D2; k++) acc += gv[k] * Wg[(long long)k * HH + h];
  pre[((long long)t * 4 + gate) * HH + h] = acc;
}

// ---------------- serial LSTM recurrence (single block, h in LDS) ----------------
__global__ __launch_bounds__(512) void gl_lstm(
    const float* __restrict__ pre,
    const float* __restrict__ W_hi, const float* __restrict__ W_hf,
    const float* __restrict__ W_hg, const float* __restrict__ W_ho,
    float* __restrict__ out) {
  __shared__ float hs[HH];
  int j = threadIdx.x;
  hs[j] = 0.0f;
  float c = 0.0f;
  __syncthreads();
  for (int t = 0; t < TT; t++) {
    const float* p = pre + (long long)t * 4 * HH;
    float ai = p[j], af = p[HH + j], ag = p[2 * HH + j], ao = p[3 * HH + j];
    for (int k = 0; k < HH; k++) {
      float h = hs[k];
      ai += h * W_hi[(long long)k * HH + j];
      af += h * W_hf[(long long)k * HH + j];
      ag += h * W_hg[(long long)k * HH + j];
      ao += h * W_ho[(long long)k * HH + j];
    }
    float ig = 1.0f / (1.0f + expf(-ai));
    float fg = 1.0f / (1.0f + expf(-af));
    float gt = tanhf(ag);
    float og = 1.0f / (1.0f + expf(-ao));
    c = fg * c + ig * gt;
    float hn = og * tanhf(c);
    out[(long long)t * HH + j] = hn;
    __syncthreads();
    hs[j] = hn;
    __syncthreads();
  }
  out[(long long)TT * HH + j] = hs[j];        // final h
  out[(long long)TT * HH + HH + j] = c;       // final c
}

// ---------------- host ----------------
extern "C" void kernel_launch(void* const* d_in, const int* in_sizes, int n_in,
                              void* d_out, int out_size, void* d_ws, size_t ws_size,
                              hipStream_t stream) {
  const float* inp     = (const float*)d_in[0];
  const float* x_nodes = (const float*)d_in[1];
  const int*   eidx    = (const int*)d_in[2];
  const float* W1  = (const float*)d_in[3];
  const float* b1  = (const float*)d_in[4];
  const float* W2  = (const float*)d_in[5];
  const float* b2  = (const float*)d_in[6];
  const float* Wfc = (const float*)d_in[7];
  const float* bfc = (const float*)d_in[8];
  const float* W_ii = (const float*)d_in[9];
  const float* W_gi = (const float*)d_in[10];
  const float* W_hi = (const float*)d_in[11];
  const float* b_i  = (const float*)d_in[12];
  const float* W_if = (const float*)d_in[13];
  const float* W_gf = (const float*)d_in[14];
  const float* W_hf = (const float*)d_in[15];
  const float* b_f  = (const float*)d_in[16];
  const float* W_ig = (const float*)d_in[17];
  const float* W_gg = (const float*)d_in[18];
  const float* W_hg = (const float*)d_in[19];
  const float* b_g  = (const float*)d_in[20];
  const float* W_io = (const float*)d_in[21];
  const float* W_go = (const float*)d_in[22];
  const float* W_ho = (const float*)d_in[23];
  const float* b_o  = (const float*)d_in[24];
  float* out = (float*)d_out;

  // workspace layout (floats)
  float* w = (float*)d_ws;
  float* deg  = w;                          // T*N              = 98304
  float* gtmp = deg  + TT * NN;             // T*56             = 5376
  float* gv   = gtmp + TT * D2;             // T*56             = 5376
  float* pre  = gv   + TT * D2;             // T*4*H            = 196608
  float* bufB = pre  + TT * 4 * HH;         // T*N*G            = 12582912
  float* bufA = bufB + (long long)TT * NN * GG; // T*N*G        = 12582912

  const int big = TT * NN * GG;             // 12,582,912

  // 1) zero deg, gtmp, accumulation buffer
  gl_zero_f32<<<(TT * NN + 255) / 256, 256, 0, stream>>>(deg, TT * NN);
  gl_zero_f32<<<(TT * D2 + 255) / 256, 256, 0, stream>>>(gtmp, TT * D2);
  gl_zero_f32<<<(big + 255) / 256, 256, 0, stream>>>(bufB, big);

  // 2) degree + dinv
  gl_deg_count<<<dim3(EE / 256, TT), 256, 0, stream>>>(eidx, deg);
  gl_dinv<<<(TT * NN + 255) / 256, 256, 0, stream>>>(deg);

  // 3) GEMM1: xw1 = x_nodes @ W1   (K = 1024)
  gl_gemm_wmma<<<dim3(NN / 128, TT), 256, 0, stream>>>(
      x_nodes, (long long)NN * NN, NN, W1, bufA, (long long)NN * GG, NN);

  // 4) scatter1 + bias + relu -> xh1 (bufB)
  gl_scatter<<<dim3(ETOT / 8, TT), 256, 0, stream>>>(eidx, deg, bufA, bufB);
  gl_bias_act<<<(big + 255) / 256, 256, 0, stream>>>(bufB, b1, 1);

  // 5) GEMM2: xw2 = xh1 @ W2   (K = 128), reuse bufA
  gl_gemm_wmma<<<dim3(NN / 128, TT), 256, 0, stream>>>(
      bufB, (long long)NN * GG, GG, W2, bufA, (long long)NN * GG, GG);

  // 6) scatter2 + bias -> xh2 (bufB, re-zeroed)
  gl_zero_f32<<<(big + 255) / 256, 256, 0, stream>>>(bufB, big);
  gl_scatter<<<dim3(ETOT / 8, TT), 256, 0, stream>>>(eidx, deg, bufA, bufB);
  gl_bias_act<<<(big + 255) / 256, 256, 0, stream>>>(bufB, b2, 0);

  // 7) FC + relu -> g
  gl_fc<<<dim3(TT, 16), 224, 0, stream>>>(bufB, Wfc, gtmp);
  gl_g_act<<<(TT * D2 + 63) / 64, 64, 0, stream>>>(gtmp, bfc, gv);

  // 8) pre-gates
  gl_pregate<<<dim3(TT, 4), 512, 0, stream>>>(
      inp, gv, W_ii, W_gi, b_i, W_if, W_gf, b_f,
      W_ig, W_gg, b_g, W_io, W_go, b_o, pre);

  // 9) serial LSTM
  gl_lstm<<<1, 512, 0, stream>>>(pre, W_hi, W_hf, W_hg, W_ho, out);
}